// ConvNd_21148418965776
// MI455X (gfx1250) — compile-verified
//
#include <hip/hip_runtime.h>

typedef float v2f __attribute__((ext_vector_type(2)));
typedef float v8f __attribute__((ext_vector_type(8)));

#define SDIM 64
#define NCIN 16
#define NCOUT 16
#define NTAP 27   // 3*3*3

// One block = 4 waves = one (b,d,h) output row (64 wide); each wave owns 16 w's.
// D(16x16: w x cout) += A(16x4: w x cin) * B(4x16: cin x cout), looped over 27 taps
// and 4 cin-groups => 108 v_wmma_f32_16x16x4_f32 per wave, all math in f32.
// LDS layouts are interleaved so each WMMA operand is ONE ds_load_b64 with an
// immediate offset from a single per-lane base address (compiler pairs them
// into ds_load_2addr_b64). Staging uses b128 global loads; halo zeroed once.
__launch_bounds__(128, 4)
__global__ void conv3d_wmma_f32(const float* __restrict__ x,
                                const float* __restrict__ w,
                                const float* __restrict__ bias,
                                float* __restrict__ out) {
    // Weights: wl[(tap*16 + cout)*16 + cin]   (6912 floats = 27648 B)
    __shared__ __align__(16) float wl[NTAP * NCOUT * NCIN];
    // x slab for one cin-group, cin interleaved innermost:
    // xs4[((dz*3+hz)*66 + wz)*4 + cin_loc], wz = w+1 (halo zeroed)  (2376 floats = 9504 B)
    __shared__ __align__(16) float xs4[9 * 66 * 4];

    const int tid  = threadIdx.x;
    const int lane = tid & 31;
    const int wv   = tid >> 5;        // wave id 0..3
    const int Lm   = lane & 15;       // M index (w) for A, N index (cout) for B/D
    const int half = lane >> 4;       // selects K pair {0,1} vs {2,3}
    const int cl   = half << 1;       // 0 or 2
    const int w0   = wv * 16;

    const int blk = blockIdx.x;       // b*4096 + d*64 + h
    const int b   = blk >> 12;
    const int d   = (blk >> 6) & 63;
    const int h   = blk & 63;

    // One-time weight transpose OIDHW -> [tap][cout][cin] in LDS.
    for (int idx = tid; idx < NCOUT * NCIN * NTAP; idx += 128) {
        int co   = idx / (NCIN * NTAP);
        int rest = idx - co * (NCIN * NTAP);
        int ci   = rest / NTAP;
        int tap  = rest - ci * NTAP;
        wl[(tap * NCOUT + co) * NCIN + ci] = w[idx];
    }
    // One-time halo zeroing: wz==0 and wz==65 are padding for every group.
    for (int t = tid; t < 9 * 4 * 2; t += 128) {
        int rr   = t >> 1;                  // row*4 + cil
        int side = t & 1;
        int row  = rr >> 2;
        int cil  = rr & 3;
        xs4[(row * 66 + side * 65) * 4 + cil] = 0.0f;
    }

    // Per-lane fragment base addresses (float indices); all taps are immediate offsets.
    const int abase = (w0 + Lm) * 4 + cl;        // + ((dz*3+hz)*66 + kw)*4
    v8f acc = {};

    for (int g = 0; g < 4; ++g) {
        __syncthreads();  // xs4 WAR across groups (and wl/halo RAW via barrier below on g=0)

        // Stage 4 cin's worth of x interior rows: 36 rows x 16 float4 (b128 loads).
        for (int t = tid; t < 36 * 16; t += 128) {
            int rr  = t >> 4;                // cil*9 + row
            int q   = t & 15;                // float4 index along w
            int cil = rr / 9;
            int row = rr - cil * 9;          // dz*3 + hz
            int zd  = d + row / 3 - 1;
            int zh  = h + (row % 3) - 1;
            int ci  = (g << 2) + cil;
            float4 v = {0.0f, 0.0f, 0.0f, 0.0f};
            if ((unsigned)zd < 64u && (unsigned)zh < 64u)
                v = *(const float4*)&x[(((b * NCIN + ci) * SDIM + zd) * SDIM + zh) * SDIM
                                       + (q << 2)];
            int base = (row * 66 + 1 + (q << 2)) * 4 + cil;   // wz = 1 + 4q
            xs4[base]      = v.x;
            xs4[base + 4]  = v.y;
            xs4[base + 8]  = v.z;
            xs4[base + 12] = v.w;
        }
        __syncthreads();

        // Pull next group's first slab line toward the WGP while the matrix pipe runs.
        if (g < 3) {
            int rr  = tid >> 4;
            int q   = tid & 15;
            int cil = rr / 9;
            int row = rr - cil * 9;
            int zd  = d + row / 3 - 1;
            int zh  = h + (row % 3) - 1;
            int ci  = ((g + 1) << 2) + cil;
            if ((unsigned)zd < 64u && (unsigned)zh < 64u)
                __builtin_prefetch(&x[(((b * NCIN + ci) * SDIM + zd) * SDIM + zh) * SDIM
                                      + (q << 2)], 0, 3);
        }

        const int bbase = Lm * NCIN + (g << 2) + cl;   // + tap*256

        // 27 taps, one K=4 cin-group per WMMA; uniform flow => EXEC all-ones.
        #pragma unroll
        for (int kd = 0; kd < 3; ++kd) {
            #pragma unroll
            for (int kh = 0; kh < 3; ++kh) {
                const int arow = abase + (kd * 3 + kh) * 66 * 4;
                #pragma unroll
                for (int kw = 0; kw < 3; ++kw) {
                    const int tap = kd * 9 + kh * 3 + kw;
                    // A: lane (M=Lm) holds K = cl,cl+1  -> one ds_load_b64
                    v2f a  = *(const v2f*)(xs4 + arow + kw * 4);
                    // B: lane (N=Lm) holds K = cl,cl+1  -> one ds_load_b64
                    v2f bm = *(const v2f*)(wl + bbase + tap * (NCOUT * NCIN));
                    acc = __builtin_amdgcn_wmma_f32_16x16x4_f32(
                        false, a, false, bm, (short)0, acc, false, false);
                }
            }
        }
    }

    // Epilogue: D VGPR r <-> w = w0 + 8*half + r. Two b128 stores per lane.
    const float bv = bias[Lm];
    float* op = out + ((((long)b * NCOUT + Lm) * SDIM + d) * SDIM + h) * SDIM
                    + w0 + 8 * half;
    float4 v0 = {acc[0] + bv, acc[1] + bv, acc[2] + bv, acc[3] + bv};
    float4 v1 = {acc[4] + bv, acc[5] + bv, acc[6] + bv, acc[7] + bv};
    *(float4*)(op)     = v0;
    *(float4*)(op + 4) = v1;
}

extern "C" void kernel_launch(void* const* d_in, const int* in_sizes, int n_in,
                              void* d_out, int out_size, void* d_ws, size_t ws_size,
                              hipStream_t stream) {
    const float* x    = (const float*)d_in[0];
    const float* wgt  = (const float*)d_in[1];
    const float* bias = (const float*)d_in[2];
    float* out        = (float*)d_out;
    (void)in_sizes; (void)n_in; (void)out_size; (void)d_ws; (void)ws_size;

    dim3 grid(4 * SDIM * SDIM);   // one block per (b, d, h) row
    conv3d_wmma_f32<<<grid, 128, 0, stream>>>(x, wgt, bias, out);
}